// VQEmbedding_8529805049925
// MI455X (gfx1250) — compile-verified
//
#include <hip/hip_runtime.h>

#define NROWS   (16 * 4096)        // 65536 vectors
#define DIM     64
#define NCODES  1024
#define TILE_M  16
#define NTILES  (NROWS / TILE_M)   // 4096 row tiles

typedef __attribute__((ext_vector_type(2))) float v2f;
typedef __attribute__((ext_vector_type(8))) float v8f;

// d_out layout: [quantized_ste: NROWS*DIM f32][commitment_loss: 1][codebook_loss: 1]
__global__ void vq_zero_losses(float* __restrict__ out) {
    if (threadIdx.x == 0) {
        out[NROWS * DIM + 0] = 0.0f;
        out[NROWS * DIM + 1] = 0.0f;
    }
}

__global__ __launch_bounds__(256)
void vq_wmma_kernel(const float* __restrict__ h,
                    const float* __restrict__ codebook,
                    float* __restrict__ out) {
    // Dynamic LDS: full codebook (256 KB) + half squared norms (4 KB) = 260 KB
    extern __shared__ float smem[];
    float* cb  = smem;                 // [NCODES][DIM]
    float* c2h = smem + NCODES * DIM;  // [NCODES] = 0.5 * ||c||^2

    const int tid = threadIdx.x;

    // ---- Stage codebook into LDS (float4 granularity) ----
    {
        const float4* src = (const float4*)codebook;
        float4* dst = (float4*)cb;
        for (int i = tid; i < (NCODES * DIM) / 4; i += blockDim.x)
            dst[i] = src[i];
    }
    __syncthreads();

    // ---- Precompute half squared norms from LDS ----
    for (int c = tid; c < NCODES; c += blockDim.x) {
        const float* row = cb + c * DIM;
        float s = 0.f;
        #pragma unroll
        for (int k = 0; k < DIM; ++k) s += row[k] * row[k];
        c2h[c] = 0.5f * s;
    }
    __syncthreads();

    const int lane = tid & 31;
    const int n    = lane & 15;   // column / A-row index within tile
    const int hi   = lane >> 4;   // half-wave select (K pair / M+8)
    const int wavesPerBlock = blockDim.x >> 5;
    const int wave  = blockIdx.x * wavesPerBlock + (tid >> 5);
    const int nWave = gridDim.x * wavesPerBlock;

    float lossSum = 0.f;

    for (int tile = wave; tile < NTILES; tile += nWave) {
        const int row0 = tile * TILE_M;

        // Prefetch next tile's A panel rows (global_prefetch_b8)
        const int nextTile = tile + nWave;
        if (nextTile < NTILES)
            __builtin_prefetch(h + ((size_t)nextTile * TILE_M + n) * DIM, 0, 1);

        // ---- Load A fragments: f32 16x4, lanes 0-15 K={4kg,4kg+1}, 16-31 K={4kg+2,4kg+3}
        const float* hrow = h + (size_t)(row0 + n) * DIM;
        v2f a[16];
        #pragma unroll
        for (int kg = 0; kg < 16; ++kg)
            a[kg] = *(const v2f*)(hrow + 4 * kg + 2 * hi);

        float best[8];
        int   bidx[8];
        #pragma unroll
        for (int r = 0; r < 8; ++r) { best[r] = -3.402823466e38f; bidx[r] = 0; }

        // ---- Sweep all code tiles: 16x16 score tile, K=64 via 16 chained WMMAs
        for (int ct = 0; ct < NCODES / 16; ++ct) {
            const int cbase = ct * 16;
            const float* brow = cb + (cbase + n) * DIM;  // B fragment from LDS
            v8f acc = {0.f, 0.f, 0.f, 0.f, 0.f, 0.f, 0.f, 0.f};
            #pragma unroll
            for (int kg = 0; kg < 16; ++kg) {
                v2f b = *(const v2f*)(brow + 4 * kg + 2 * hi);
                acc = __builtin_amdgcn_wmma_f32_16x16x4_f32(
                    /*neg_a=*/false, a[kg], /*neg_b=*/false, b,
                    /*c_mod=*/(short)0, acc, /*reuse_a=*/false, /*reuse_b=*/false);
            }
            // score = h·c - 0.5||c||^2 ; argmax score == argmin squared distance
            const float ch   = c2h[cbase + n];
            const int   code = cbase + n;
            #pragma unroll
            for (int r = 0; r < 8; ++r) {
                const float s = acc[r] - ch;
                if (s > best[r]) { best[r] = s; bidx[r] = code; }
            }
        }

        // ---- Argmax reduction across the 16 lanes of each half-wave
        #pragma unroll
        for (int r = 0; r < 8; ++r) {
            #pragma unroll
            for (int off = 8; off >= 1; off >>= 1) {
                const float ov = __shfl_xor(best[r], off, 32);
                const int   oi = __shfl_xor(bidx[r], off, 32);
                if (ov > best[r] || (ov == best[r] && oi < bidx[r])) {
                    best[r] = ov; bidx[r] = oi;
                }
            }
        }

        // ---- Gather quantized rows from LDS, store, accumulate MSE partial
        // lanes 0-15 handle rows row0+0..7 (r), lanes 16-31 rows row0+8..15
        #pragma unroll
        for (int r = 0; r < 8; ++r) {
            const int rowAbs = row0 + r + 8 * hi;
            const int code   = bidx[r];   // uniform within half-wave
            const float4 q  = *(const float4*)(cb + code * DIM + 4 * n);
            const float4 hv = *(const float4*)(h + (size_t)rowAbs * DIM + 4 * n);
            *(float4*)(out + (size_t)rowAbs * DIM + 4 * n) = q;
            const float dx = hv.x - q.x, dy = hv.y - q.y;
            const float dz = hv.z - q.z, dw = hv.w - q.w;
            lossSum += dx * dx + dy * dy + dz * dz + dw * dw;
        }
    }

    // ---- Loss: wave reduction then global atomic accumulation
    #pragma unroll
    for (int off = 16; off >= 1; off >>= 1)
        lossSum += __shfl_xor(lossSum, off, 32);
    if (lane == 0) {
        const float v = lossSum * (1.0f / (float)(NROWS * DIM));
        atomicAdd(out + NROWS * DIM + 0, 0.25f * v);  // commitment_loss
        atomicAdd(out + NROWS * DIM + 1, v);          // codebook_loss
    }
}

extern "C" void kernel_launch(void* const* d_in, const int* in_sizes, int n_in,
                              void* d_out, int out_size, void* d_ws, size_t ws_size,
                              hipStream_t stream) {
    const float* h        = (const float*)d_in[0];  // [16,4096,64] f32
    const float* codebook = (const float*)d_in[1];  // [1024,64] f32
    float* out = (float*)d_out;

    vq_zero_losses<<<1, 32, 0, stream>>>(out);

    const size_t shmem = (size_t)(NCODES * DIM + NCODES) * sizeof(float); // 260 KB < 320 KB
    // 256 blocks x 8 waves = 2048 waves; 4096 row tiles -> 2 tiles/wave
    vq_wmma_kernel<<<256, 256, shmem, stream>>>(h, codebook, out);
}